// GraphWholeMessage_54107997995649
// MI455X (gfx1250) — compile-verified
//
#include <hip/hip_runtime.h>
#include <hip/hip_bf16.h>

#define BS  16
#define T   12
#define LAG 8
#define NN  512
#define EMB 64
#define KC  2   // K graph channels

typedef float v2f __attribute__((ext_vector_type(2)));
typedef float v8f __attribute__((ext_vector_type(8)));

// CDNA5 fp32 WMMA: D(16x16) = A(16x4) * B(4x16) + C
// 8-arg form: (neg_a, A, neg_b, B, c_mod, C, reuse_a, reuse_b)
__device__ __forceinline__ v8f wmma4(v2f a, v2f b, v8f c) {
  return __builtin_amdgcn_wmma_f32_16x16x4_f32(false, a, false, b, (short)0, c,
                                               false, false);
}

// ---------------------------------------------------------------------------
// hidden init: lp = 2*label-1, keep only last occurrence of each skill
// ---------------------------------------------------------------------------
__global__ void k_init_hidden(const int* __restrict__ skills,
                              const int* __restrict__ labels,
                              const float* __restrict__ node_emb,
                              float* __restrict__ hidden) {
  int b = blockIdx.x;
  int e = threadIdx.x;   // 0..63
  for (int l = 0; l < LAG; ++l) {
    int s = skills[b * T + l];
    bool is_last = true;
    for (int l2 = l + 1; l2 < LAG; ++l2)
      if (skills[b * T + l2] == s) { is_last = false; break; }
    if (!is_last) continue;
    float lp = (labels[b * T + l] == 0) ? -1.0f : 1.0f;
    hidden[((size_t)b * NN + s) * EMB + e] = node_emb[s * EMB + e] * lp;
  }
}

// ---------------------------------------------------------------------------
// adj2rows[g][slot][n] = (adj[g] @ adj[g])[skills[b,slot], n] / N
// Only the 12 rows indexed by skills are ever consumed -> 16x512x512 GEMM
// per batch instead of 512x512x512.  One block = 4 waves = 64 N-columns.
// ---------------------------------------------------------------------------
__global__ void k_adj2rows(const float* __restrict__ adj,
                           const int* __restrict__ skills,
                           float* __restrict__ adj2rows) {
  __shared__ float Atile[16 * NN];           // 32 KB of the 320 KB WGP LDS
  int g = blockIdx.y;                        // k*BS + b
  int b = g & (BS - 1);
  const float* Ab = adj + (size_t)g * NN * NN;
  int tid = threadIdx.x;

  for (int i = tid; i < 16 * NN; i += 128) {
    int slot = i >> 9, kc = i & (NN - 1);
    float v = 0.0f;
    if (slot < T) {
      int srow = skills[b * T + slot];
      v = Ab[(size_t)srow * NN + kc];
    }
    Atile[i] = v;
  }
  __syncthreads();

  int lane = tid & 31, wave = tid >> 5;
  int col = lane & 15, grp = lane >> 4;      // grp selects K-pair / M-half
  int n0 = blockIdx.x * 64 + wave * 16;

  v8f acc = {};
  for (int k = 0; k < NN; k += 4) {
    if ((k & 31) == 0 && k + 32 < NN)        // uniform; EXEC untouched
      __builtin_prefetch(Ab + (size_t)(k + 32) * NN + n0, 0, 1);
    int ka = k + 2 * grp;
    v2f a  = { Atile[(lane & 15) * NN + ka], Atile[(lane & 15) * NN + ka + 1] };
    v2f bb = { Ab[(size_t)ka * NN + n0 + col],
               Ab[(size_t)(ka + 1) * NN + n0 + col] };
    acc = wmma4(a, bb, acc);
  }
  const float inv_n = 1.0f / (float)NN;
  for (int i = 0; i < 8; ++i) {
    int slot = i + 8 * grp;
    if (slot < T)
      adj2rows[((size_t)g * T + slot) * NN + n0 + col] = acc[i] * inv_n;
  }
}

// ---------------------------------------------------------------------------
// per-t: hh gather, U = hh @ W1_top + b1  (128 rows only), time weights
// ---------------------------------------------------------------------------
__global__ void k_prep(int t, const int* __restrict__ skills,
                       const int* __restrict__ times,
                       const float* __restrict__ hidden,
                       const float* __restrict__ msg_w1,
                       const float* __restrict__ msg_b1,
                       float* __restrict__ U, float* __restrict__ wvec) {
  __shared__ float hhl[EMB];
  int b = blockIdx.x >> 3;
  int l = blockIdx.x & 7;
  int e = threadIdx.x;
  int s = skills[b * T + (t - LAG + l)];
  hhl[e] = hidden[((size_t)b * NN + s) * EMB + e];
  __syncthreads();
  float acc = msg_b1[e];
  for (int k = 0; k < EMB; ++k) acc += hhl[k] * msg_w1[k * EMB + e];
  U[(b * LAG + l) * EMB + e] = acc;
  if (e == 0) {
    float dt  = fabsf((float)times[b * T + (t - LAG + l)] -
                      (float)times[b * T + t]);
    float dtw = logf(dt + 1e-10f) / logf(5.0f);
    wvec[b * LAG + l] = expf(-dtw);
  }
}

// ---------------------------------------------------------------------------
// Hb = hidden(8192x64) @ W1_bot(64x64)   (WMMA)
// ---------------------------------------------------------------------------
__global__ void k_gemm_hb(const float* __restrict__ hidden,
                          const float* __restrict__ msg_w1,
                          float* __restrict__ Hb) {
  int tid = threadIdx.x;
  int lane = tid & 31, wave = tid >> 5;
  int col = lane & 15, grp = lane >> 4;
  int r0 = blockIdx.x * 16;
  int n0 = wave * 16;
  const float* W = msg_w1 + EMB * EMB;       // bottom half rows 64..127
  v8f acc = {};
  for (int k = 0; k < EMB; k += 4) {
    int ka  = k + 2 * grp;
    int row = r0 + (lane & 15);
    v2f a  = { hidden[(size_t)row * EMB + ka],
               hidden[(size_t)row * EMB + ka + 1] };
    v2f bb = { W[ka * EMB + n0 + col], W[(ka + 1) * EMB + n0 + col] };
    acc = wmma4(a, bb, acc);
  }
  for (int i = 0; i < 8; ++i) {
    int row = r0 + i + 8 * grp;
    Hb[(size_t)row * EMB + n0 + col] = acc[i];
  }
}

// ---------------------------------------------------------------------------
// msg2 = tanh( tanh(U[b,l]+Hb[b,n]) @ W2 + b2 )   (WMMA, A-tile in LDS)
// row r = (b*LAG+l)*NN + n
// ---------------------------------------------------------------------------
__global__ void k_msg2(const float* __restrict__ U,
                       const float* __restrict__ Hb,
                       const float* __restrict__ msg_w2,
                       const float* __restrict__ msg_b2,
                       float* __restrict__ msg2) {
  __shared__ float Atile[16 * EMB];
  int tid = threadIdx.x;
  int r0 = blockIdx.x * 16;
  for (int i = tid; i < 16 * EMB; i += 128) {
    int rl = i >> 6, kc = i & 63;
    int r = r0 + rl;
    int b = r >> 12, l = (r >> 9) & 7, n = r & 511;
    Atile[i] = tanhf(U[(b * LAG + l) * EMB + kc] +
                     Hb[((size_t)b * NN + n) * EMB + kc]);
  }
  __syncthreads();
  int lane = tid & 31, wave = tid >> 5;
  int col = lane & 15, grp = lane >> 4;
  int n0 = wave * 16;
  v8f acc = {};
  for (int k = 0; k < EMB; k += 4) {
    int ka = k + 2 * grp;
    v2f a  = { Atile[(lane & 15) * EMB + ka], Atile[(lane & 15) * EMB + ka + 1] };
    v2f bb = { msg_w2[ka * EMB + n0 + col], msg_w2[(ka + 1) * EMB + n0 + col] };
    acc = wmma4(a, bb, acc);
  }
  for (int i = 0; i < 8; ++i) {
    int row = r0 + i + 8 * grp;
    msg2[(size_t)row * EMB + n0 + col] = tanhf(acc[i] + msg_b2[n0 + col]);
  }
}

// ---------------------------------------------------------------------------
// agg[b,k,n,e] = sum_l (adj[k,b,sh,n] + adj2rows[k,b,slot,n]) * w[b,l] * msg2
// hidden[b,n,e] = mean over k
// ---------------------------------------------------------------------------
__global__ void k_agg(int t, const int* __restrict__ skills,
                      const float* __restrict__ adj,
                      const float* __restrict__ adj2rows,
                      const float* __restrict__ wvec,
                      const float* __restrict__ msg2,
                      float* __restrict__ agg, float* __restrict__ hidden) {
  __shared__ float cs[2][LAG];
  __shared__ float ws[LAG];
  int bn = blockIdx.x;
  int b = bn >> 9;
  int n = bn & 511;
  int e = threadIdx.x;
  if (e < 16) {
    int kidx = e >> 3, l = e & 7;
    int slot = t - LAG + l;
    int s = skills[b * T + slot];
    int g = kidx * BS + b;
    cs[kidx][l] = adj[((size_t)g * NN + s) * NN + n] +
                  adj2rows[((size_t)g * T + slot) * NN + n];
    if (kidx == 0) ws[l] = wvec[b * LAG + l];
  }
  __syncthreads();
  float a0 = 0.0f, a1 = 0.0f;
  for (int l = 0; l < LAG; ++l) {
    float m = msg2[(((size_t)(b * LAG + l)) * NN + n) * EMB + e] * ws[l];
    a0 += cs[0][l] * m;
    a1 += cs[1][l] * m;
  }
  agg[(((size_t)b * KC + 0) * NN + n) * EMB + e] = a0;
  agg[(((size_t)b * KC + 1) * NN + n) * EMB + e] = a1;
  hidden[((size_t)b * NN + n) * EMB + e] = 0.5f * (a0 + a1);
}

// ---------------------------------------------------------------------------
// mlp_final + softmax only at n = skills[b,t]  (32 rows total)
// ---------------------------------------------------------------------------
__global__ void k_final(int t, const int* __restrict__ skills,
                        const float* __restrict__ agg,
                        const float* f_w0, const float* f_b0,
                        const float* f_g0, const float* f_be0,
                        const float* f_w1, const float* f_b1,
                        const float* f_g1, const float* f_be1,
                        const float* f_w2, const float* f_b2,
                        float* __restrict__ out) {
  __shared__ float x[EMB], h1[EMB], tbuf[EMB], h2[EMB];
  int bk = blockIdx.x;               // b*2 + k
  int b = bk >> 1, kk = bk & 1;
  int e = threadIdx.x;
  int st = skills[b * T + t];
  x[e] = agg[(((size_t)b * KC + kk) * NN + st) * EMB + e];
  __syncthreads();
  // layer 0: x @ f_w0 + b0 -> LN -> leaky_relu
  float acc = f_b0[e];
  for (int k2 = 0; k2 < EMB; ++k2) acc += x[k2] * f_w0[k2 * EMB + e];
  tbuf[e] = acc;
  __syncthreads();
  float mu = 0.0f, var = 0.0f;
  for (int k2 = 0; k2 < EMB; ++k2) mu += tbuf[k2];
  mu *= (1.0f / EMB);
  for (int k2 = 0; k2 < EMB; ++k2) { float d = tbuf[k2] - mu; var += d * d; }
  var *= (1.0f / EMB);
  float v = (tbuf[e] - mu) * rsqrtf(var + 1e-5f) * f_g0[e] + f_be0[e];
  h1[e] = (v >= 0.0f) ? v : 0.01f * v;
  __syncthreads();
  // layer 1: h1 @ f_w1 + b1 -> LN -> leaky_relu, residual
  acc = f_b1[e];
  for (int k2 = 0; k2 < EMB; ++k2) acc += h1[k2] * f_w1[k2 * EMB + e];
  tbuf[e] = acc;
  __syncthreads();
  mu = 0.0f; var = 0.0f;
  for (int k2 = 0; k2 < EMB; ++k2) mu += tbuf[k2];
  mu *= (1.0f / EMB);
  for (int k2 = 0; k2 < EMB; ++k2) { float d = tbuf[k2] - mu; var += d * d; }
  var *= (1.0f / EMB);
  v = (tbuf[e] - mu) * rsqrtf(var + 1e-5f) * f_g1[e] + f_be1[e];
  v = (v >= 0.0f) ? v : 0.01f * v;
  h2[e] = h1[e] + v;
  __syncthreads();
  if (e == 0) {
    float l0 = f_b2[0], l1 = f_b2[1];
    for (int k2 = 0; k2 < EMB; ++k2) {
      l0 += h2[k2] * f_w2[k2 * 2 + 0];
      l1 += h2[k2] * f_w2[k2 * 2 + 1];
    }
    float mx = fmaxf(l0, l1);
    float e0 = expf(l0 - mx), e1 = expf(l1 - mx);
    out[((t - LAG) * BS + b) * KC + kk] = e1 / (e0 + e1);
  }
}

// ---------------------------------------------------------------------------
extern "C" void kernel_launch(void* const* d_in, const int* in_sizes, int n_in,
                              void* d_out, int out_size, void* d_ws,
                              size_t ws_size, hipStream_t stream) {
  (void)in_sizes; (void)n_in; (void)out_size; (void)ws_size;
  const int*   skills   = (const int*)d_in[0];
  const int*   times    = (const int*)d_in[2];
  const int*   labels   = (const int*)d_in[3];
  const float* adj      = (const float*)d_in[4];
  const float* node_emb = (const float*)d_in[5];
  const float* msg_w1   = (const float*)d_in[6];
  const float* msg_b1   = (const float*)d_in[7];
  const float* msg_w2   = (const float*)d_in[8];
  const float* msg_b2   = (const float*)d_in[9];
  const float* f_w0  = (const float*)d_in[10];
  const float* f_b0  = (const float*)d_in[11];
  const float* f_g0  = (const float*)d_in[12];
  const float* f_be0 = (const float*)d_in[13];
  const float* f_w1  = (const float*)d_in[14];
  const float* f_b1  = (const float*)d_in[15];
  const float* f_g1  = (const float*)d_in[16];
  const float* f_be1 = (const float*)d_in[17];
  const float* f_w2  = (const float*)d_in[18];
  const float* f_b2  = (const float*)d_in[19];

  float* ws       = (float*)d_ws;
  float* hidden   = ws;                         // 16*512*64      = 524288
  float* adj2rows = hidden + 524288;            // 2*16*12*512    = 196608
  float* U        = adj2rows + 196608;          // 16*8*64        = 8192
  float* Hb       = U + 8192;                   // 16*512*64      = 524288
  float* msg2     = Hb + 524288;                // 16*8*512*64    = 4194304
  float* agg      = msg2 + 4194304;             // 16*2*512*64    = 1048576
  float* wvec     = agg + 1048576;              // 16*8           = 128

  hipMemsetAsync(hidden, 0, (size_t)524288 * sizeof(float), stream);
  k_init_hidden<<<BS, EMB, 0, stream>>>(skills, labels, node_emb, hidden);
  k_adj2rows<<<dim3(8, KC * BS), 128, 0, stream>>>(adj, skills, adj2rows);

  for (int t = LAG; t < T; ++t) {
    k_prep<<<BS * LAG, EMB, 0, stream>>>(t, skills, times, hidden,
                                         msg_w1, msg_b1, U, wvec);
    k_gemm_hb<<<(BS * NN) / 16, 128, 0, stream>>>(hidden, msg_w1, Hb);
    k_msg2<<<(BS * LAG * NN) / 16, 128, 0, stream>>>(U, Hb, msg_w2, msg_b2,
                                                     msg2);
    k_agg<<<BS * NN, EMB, 0, stream>>>(t, skills, adj, adj2rows, wvec, msg2,
                                       agg, hidden);
    k_final<<<BS * KC, EMB, 0, stream>>>(t, skills, agg,
                                         f_w0, f_b0, f_g0, f_be0,
                                         f_w1, f_b1, f_g1, f_be1,
                                         f_w2, f_b2, (float*)d_out);
  }
}